// RBF_90821378441796
// MI455X (gfx1250) — compile-verified
//
#include <hip/hip_runtime.h>
#include <hip/hip_bf16.h>

// Shapes: B=1024, N=32, C=2048
// out[b,c] = sum_n v[b,n] * x[b,n,c],  where A v = phi0 (A symmetric 32x32)

typedef float v2f __attribute__((ext_vector_type(2)));
typedef float v8f __attribute__((ext_vector_type(8)));

#define NB 1024
#define NP 32
#define NC 2048

__device__ __forceinline__ float wave_max32(float v) {
    for (int o = 16; o; o >>= 1) v = fmaxf(v, __shfl_xor(v, o, 32));
    return v;
}
__device__ __forceinline__ float wave_min32(float v) {
    for (int o = 16; o; o >>= 1) v = fminf(v, __shfl_xor(v, o, 32));
    return v;
}
__device__ __forceinline__ float wave_sum32(float v) {
    for (int o = 16; o; o >>= 1) v += __shfl_xor(v, o, 32);
    return v;
}

// One wave (32 lanes) per batch: build 32x32 RBF matrix, Gaussian elimination
// with partial pivoting, back-substitution. Writes v[b][0..31] to vout.
__global__ void __launch_bounds__(32) rbf_solve_kernel(const float* __restrict__ xs,
                                                       float* __restrict__ vout) {
    __shared__ float A[NP][NP + 2];   // 33 used columns (augmented), +1 pad
    __shared__ float spy[NP], spz[NP], vv[NP];

    const int b = blockIdx.x;
    const int lane = threadIdx.x;

    const float py = xs[(size_t)b * (NP * 3) + lane * 3 + 1];
    const float pz = xs[(size_t)b * (NP * 3) + lane * 3 + 2];

    const float ey = wave_max32(py) - wave_min32(py);
    const float ez = wave_max32(pz) - wave_min32(pz);
    const float inv_eps2 = (float)NP / (ey * ez);   // 1/eps^2

    spy[lane] = py;
    spz[lane] = pz;
    __syncthreads();

    // Row `lane` of augmented matrix [A | phi0]
    #pragma unroll 4
    for (int j = 0; j < NP; ++j) {
        const float dy = py - spy[j];
        const float dz = pz - spz[j];
        A[lane][j] = sqrtf((dy * dy + dz * dz) * inv_eps2 + 1.0f);
    }
    A[lane][NP] = sqrtf((py * py + pz * pz) * inv_eps2 + 1.0f);
    __syncthreads();

    // Gaussian elimination with partial pivoting (lanes act on columns)
    for (int k = 0; k < NP; ++k) {
        // argmax_{i>=k} |A[i][k]| via shuffle reduction (deterministic tie-break)
        float pvv = (lane >= k) ? fabsf(A[lane][k]) : -1.0f;
        int pr = lane;
        for (int o = 16; o; o >>= 1) {
            const float ov = __shfl_xor(pvv, o, 32);
            const int oi = __shfl_xor(pr, o, 32);
            if (ov > pvv || (ov == pvv && oi < pr)) { pvv = ov; pr = oi; }
        }
        if (pr != k) {
            const float t0 = A[k][lane];
            A[k][lane] = A[pr][lane];
            A[pr][lane] = t0;
            if (lane == 0) {
                const float t1 = A[k][NP];
                A[k][NP] = A[pr][NP];
                A[pr][NP] = t1;
            }
        }
        __syncthreads();

        const float akk_inv = 1.0f / A[k][k];
        for (int i = k + 1; i < NP; ++i) {
            const float f = A[i][k] * akk_inv;   // uniform LDS read (before write below)
            A[i][lane] -= f * A[k][lane];
            if (lane == 0) A[i][NP] -= f * A[k][NP];
        }
        __syncthreads();
    }

    // Back substitution (upper triangular)
    for (int k = NP - 1; k >= 0; --k) {
        const float t = (lane > k) ? A[k][lane] * vv[lane] : 0.0f;
        const float s = wave_sum32(t);
        if (lane == 0) vv[k] = (A[k][NP] - s) / A[k][k];
        __syncthreads();
    }

    vout[(size_t)b * NP + lane] = vv[lane];
}

// out[b,c] = sum_n v[b,n] * x[b,n,c] via V_WMMA_F32_16X16X4_F32.
// A-operand: v replicated across all 16 M rows (lane half selects K pair {0,1}/{2,3}).
// B-operand: 16-column tile of x; lane = N column, half-wave = K row pair -> loads
// are two contiguous 64B segments per instruction (coalesced main stream).
// x is 256MB streamed exactly once (> 192MB L2), so all x loads use the
// non-temporal hint; out is write-once -> non-temporal stores.
// Every lane's acc[0] holds out[c0 + lane%16]; low half-wave stores it.
__global__ void __launch_bounds__(256) rbf_einsum_wmma_kernel(const float* __restrict__ x,
                                                              const float* __restrict__ v,
                                                              float* __restrict__ out) {
    const int b = blockIdx.x;                 // 0..1023
    const int wave = threadIdx.x >> 5;        // 0..7
    const int lane = threadIdx.x & 31;
    const int half = lane >> 4;               // 0 or 1 -> K offset 0 / 2
    const int cn = lane & 15;                 // WMMA N index = column within tile

    const float* __restrict__ xb = x + (size_t)b * NP * NC;
    const float* __restrict__ vb = v + (size_t)b * NP;

    // Preload A-operand pairs: a[kk] covers K = 4*kk + {2*half, 2*half+1}
    v2f a[8];
    #pragma unroll
    for (int kk = 0; kk < 8; ++kk) {
        a[kk].x = vb[4 * kk + 2 * half];
        a[kk].y = vb[4 * kk + 2 * half + 1];
    }

    const int c_base = wave * 256;            // each wave handles 256 columns
    const float* __restrict__ xl = xb + (size_t)(2 * half) * NC + cn;

    for (int t = 0; t < 16; ++t) {
        const int c0 = c_base + t * 16;
        v8f acc = {};
        #pragma unroll
        for (int kk = 0; kk < 8; ++kk) {
            v2f bm;
            bm.x = __builtin_nontemporal_load(&xl[(size_t)(4 * kk) * NC + c0]);      // K = 4kk + 2*half
            bm.y = __builtin_nontemporal_load(&xl[(size_t)(4 * kk + 1) * NC + c0]);  // K = 4kk + 2*half + 1
            acc = __builtin_amdgcn_wmma_f32_16x16x4_f32(
                /*neg_a=*/false, a[kk], /*neg_b=*/false, bm,
                /*c_mod=*/(short)0, acc, /*reuse_a=*/false, /*reuse_b=*/false);
        }
        if (lane < 16) __builtin_nontemporal_store(acc[0], &out[(size_t)b * NC + c0 + lane]);
    }
}

extern "C" void kernel_launch(void* const* d_in, const int* in_sizes, int n_in,
                              void* d_out, int out_size, void* d_ws, size_t ws_size,
                              hipStream_t stream) {
    (void)in_sizes; (void)n_in; (void)out_size; (void)ws_size;
    const float* xs = (const float*)d_in[0];   // (1024, 32, 3)
    const float* x  = (const float*)d_in[1];   // (1024, 32, 2048)
    // d_in[2] (inputs), d_in[3] (train), d_in[4] (stage) are unused by the math.
    float* out = (float*)d_out;                // (1024, 2048)
    float* v   = (float*)d_ws;                 // 1024*32 floats = 128 KB scratch

    rbf_solve_kernel<<<dim3(NB), dim3(32), 0, stream>>>(xs, v);
    rbf_einsum_wmma_kernel<<<dim3(NB), dim3(256), 0, stream>>>(x, v, out);
}